// Integrator_70257075028065
// MI455X (gfx1250) — compile-verified
//
#include <hip/hip_runtime.h>

typedef float v2f __attribute__((ext_vector_type(2)));
typedef float v8f __attribute__((ext_vector_type(8)));

#define T_LEN        1048576          // time axis length (fixed by reference)
#define SEG          4096             // elements per wave-segment in pass 1
#define TILES_PER_SEG (SEG / 256)     // 16 WMMA tiles per segment
#define SEGS_PER_ROW (T_LEN / SEG)    // 256 segments per row

// ---------------------------------------------------------------------------
// Pass 1: one wave32 per 4096-element segment. Each 256-element tile is a
// row-major 16x16 matrix X. Full tile scan via two WMMA groups on one
// accumulator:
//   acc  = X * U        (U  = upper-tri ones  -> prefix within each 16-row)
//   acc += Ls * W       (Ls = strictly-lower-tri ones, W[k][*] = rowsum_k
//                        -> adds exclusive prefix of row sums to each row)
// Row sums are read from X*U at column N=15 (lanes 15/31 of the D layout).
// ---------------------------------------------------------------------------
__global__ __launch_bounds__(32) void seg_scan_wmma(const float* __restrict__ x,
                                                    float* __restrict__ y,
                                                    float* __restrict__ segSums) {
  const int seg  = blockIdx.x;
  const int lane = threadIdx.x;      // wave32 lane
  const int n    = lane & 15;        // N (for B/D), M (for A operands)
  const int hi   = lane >> 4;        // lane half: selects K pair

  const long long segBase = (long long)seg * SEG;

  // Constant B operand: U[k][n] = (k <= n). B layout (mirror of documented
  // 32-bit A layout): chunk kc, VGPR0 = row (4kc+2*hi), VGPR1 = next row.
  v2f bU[4];
  // Constant A operand: Ls[m][k] = (k < m), m = n. A layout: VGPR0 = K=(4kc+2hi).
  v2f aLs[4];
#pragma unroll
  for (int kc = 0; kc < 4; ++kc) {
    const int k0 = 4 * kc + 2 * hi;
    bU[kc].x  = (k0     <= n) ? 1.0f : 0.0f;
    bU[kc].y  = (k0 + 1 <= n) ? 1.0f : 0.0f;
    aLs[kc].x = (k0     <  n) ? 1.0f : 0.0f;
    aLs[kc].y = (k0 + 1 <  n) ? 1.0f : 0.0f;
  }

  float carry = 0.0f;

  for (int t = 0; t < TILES_PER_SEG; ++t) {
    const float* tx = x + segBase + t * 256;

    // A = X chunks (16x4): lane holds row M=n, VGPR0 = K=(4kc+2hi), VGPR1 = K+1.
    v2f a[4];
#pragma unroll
    for (int kc = 0; kc < 4; ++kc) {
      a[kc] = *(const v2f*)(tx + 16 * n + 4 * kc + 2 * hi);
    }

    if (t + 1 < TILES_PER_SEG) {
      __builtin_prefetch(tx + 256 + lane * 8, 0, 1);  // global_prefetch_b8
    }

    // acc = X * U : inclusive prefix along each row of 16.
    v8f acc = {};
#pragma unroll
    for (int kc = 0; kc < 4; ++kc) {
      acc = __builtin_amdgcn_wmma_f32_16x16x4_f32(
          false, a[kc], false, bU[kc], (short)0, acc, false, false);
    }

    // Row sums sit at N=15: rows 0..7 -> lane 15, rows 8..15 -> lane 31.
    float rs[16];
#pragma unroll
    for (int r = 0; r < 8; ++r) {
      rs[r]     = __shfl(acc[r], 15, 32);
      rs[r + 8] = __shfl(acc[r], 31, 32);
    }

    // W in B layout: row k of W is rs[k] broadcast across N.
    v2f w[4];
#pragma unroll
    for (int kc = 0; kc < 4; ++kc) {
      w[kc].x = hi ? rs[4 * kc + 2] : rs[4 * kc];
      w[kc].y = hi ? rs[4 * kc + 3] : rs[4 * kc + 1];
    }

    // acc += Ls * W : adds exclusive prefix of row sums to every row element.
#pragma unroll
    for (int kc = 0; kc < 4; ++kc) {
      acc = __builtin_amdgcn_wmma_f32_16x16x4_f32(
          false, aLs[kc], false, w[kc], (short)0, acc, false, false);
    }

    // Store: D layout -> element (M = r + 8*hi, N = n) at index 16*M + N.
    float* ty = y + segBase + t * 256;
#pragma unroll
    for (int r = 0; r < 8; ++r) {
      ty[16 * r + 128 * hi + n] = acc[r] + carry;
    }

    // Tile total (uniform across lanes: rs built from broadcast shuffles).
    float tot = 0.0f;
#pragma unroll
    for (int i = 0; i < 16; ++i) tot += rs[i];
    carry += tot;
  }

  if (lane == 0) segSums[seg] = carry;
}

// ---------------------------------------------------------------------------
// Pass 2: one block per row; exclusive scan of the 256 segment totals.
// ---------------------------------------------------------------------------
__global__ __launch_bounds__(SEGS_PER_ROW) void scan_seg_sums(
    const float* __restrict__ sums, float* __restrict__ offs) {
  __shared__ float s[SEGS_PER_ROW];
  const int row = blockIdx.x;
  const int t   = threadIdx.x;
  const float orig = sums[row * SEGS_PER_ROW + t];
  s[t] = orig;
  __syncthreads();
#pragma unroll
  for (int d = 1; d < SEGS_PER_ROW; d <<= 1) {
    const float v = (t >= d) ? s[t - d] : 0.0f;
    __syncthreads();
    s[t] += v;
    __syncthreads();
  }
  offs[row * SEGS_PER_ROW + t] = s[t] - orig;  // exclusive prefix
}

// ---------------------------------------------------------------------------
// Pass 3: add per-segment exclusive offset to every element (float4 wide).
// ---------------------------------------------------------------------------
__global__ __launch_bounds__(256) void add_offsets(float* __restrict__ y,
                                                   const float* __restrict__ offs) {
  const int seg = blockIdx.x;
  const float o = offs[seg];
  float4* p = (float4*)(y + (long long)seg * SEG);
  const int t = threadIdx.x;
#pragma unroll
  for (int i = 0; i < SEG / 4 / 256; ++i) {   // 4 iterations
    float4 v = p[t + i * 256];
    v.x += o; v.y += o; v.z += o; v.w += o;
    p[t + i * 256] = v;
  }
}

// ---------------------------------------------------------------------------
extern "C" void kernel_launch(void* const* d_in, const int* in_sizes, int n_in,
                              void* d_out, int out_size, void* d_ws, size_t ws_size,
                              hipStream_t stream) {
  (void)n_in; (void)out_size; (void)ws_size;
  const float* x = (const float*)d_in[0];
  float* y = (float*)d_out;

  const long long total = (long long)in_sizes[0];     // B*C*T
  const int rows = (int)(total / T_LEN);              // 32
  const int nsegTotal = rows * SEGS_PER_ROW;          // 8192

  float* sums = (float*)d_ws;                         // [nsegTotal]
  float* offs = sums + nsegTotal;                     // [nsegTotal]

  seg_scan_wmma<<<nsegTotal, 32, 0, stream>>>(x, y, sums);
  scan_seg_sums<<<rows, SEGS_PER_ROW, 0, stream>>>(sums, offs);
  add_offsets<<<nsegTotal, 256, 0, stream>>>(y, offs);
}